// GN_67250597921413
// MI455X (gfx1250) — compile-verified
//
#include <hip/hip_runtime.h>
#include <hip/hip_bf16.h>

typedef __attribute__((ext_vector_type(2))) float v2f;
typedef __attribute__((ext_vector_type(8))) float v8f;

#define GN_D   64
#define GN_LDW 66   // padded LDS row pitch (floats) -> conflict-free ds_load_b64

// ---------------- zero workspace ----------------
__global__ void gn_zero(float* __restrict__ p, long long n) {
    long long i = (long long)blockIdx.x * blockDim.x + threadIdx.x;
    long long stride = (long long)gridDim.x * blockDim.x;
    for (; i < n; i += stride) p[i] = 0.0f;
}

// ---------------- degree histogram ----------------
__global__ void gn_degree(const int* __restrict__ src, const int* __restrict__ dst,
                          float* __restrict__ deg_out, float* __restrict__ deg_in, int E) {
    int e = blockIdx.x * blockDim.x + threadIdx.x;
    if (e < E) {
        atomicAdd(&deg_out[src[e]], 1.0f);
        atomicAdd(&deg_in[dst[e]], 1.0f);
    }
}

// ---------------- deg -> rsqrt(clip(deg,1)) in place (covers both arrays) ----------------
__global__ void gn_norm(float* __restrict__ deg, int n) {
    int i = blockIdx.x * blockDim.x + threadIdx.x;
    if (i < n) {
        float d = deg[i];
        deg[i] = rsqrtf(d < 1.0f ? 1.0f : d);
    }
}

// ---------------- edge gather + scatter-add: agg[dst] += x[src] * norm_src[src] ----------------
// 16 lanes per edge, each lane moves a float4 (L2-resident working set).
__global__ void gn_scatter(const float* __restrict__ x, const int* __restrict__ src,
                           const int* __restrict__ dst, const float* __restrict__ norm_src,
                           float* __restrict__ agg, int E) {
    long long t = (long long)blockIdx.x * blockDim.x + threadIdx.x;
    int e = (int)(t >> 4);
    if (e >= E) return;
    int c4 = (int)(t & 15);
    int s = src[e];
    int d = dst[e];
    float ns = norm_src[s];
    const float4 xv = ((const float4*)(x + (size_t)s * GN_D))[c4];
    float* ap = agg + (size_t)d * GN_D + c4 * 4;
    atomicAdd(ap + 0, xv.x * ns);
    atomicAdd(ap + 1, xv.y * ns);
    atomicAdd(ap + 2, xv.z * ns);
    atomicAdd(ap + 3, xv.w * ns);
}

// ---------------- WMMA GEMM: out = (agg * norm_dst[:,None]) @ W + b ----------------
// 256 threads = 8 waves; each wave computes a 16x64 strip via V_WMMA_F32_16X16X4_F32.
// W is staged transposed into LDS (Wt[col][k], pitch 66) so the B fragment is one ds_load_b64.
__global__ void __launch_bounds__(256) gn_gemm_wmma(const float* __restrict__ agg,
                                                    const float* __restrict__ norm_dst,
                                                    const float* __restrict__ W,
                                                    const float* __restrict__ bias,
                                                    float* __restrict__ out, int n_nodes) {
    __shared__ float Wt[GN_D * GN_LDW];

    int tid = threadIdx.x;
    // stage W transposed: Wt[j*LDW + i] = W[i][j]
    for (int idx = tid; idx < GN_D * GN_D; idx += 256) {
        int i = idx >> 6;   // input-channel k
        int j = idx & 63;   // output col
        Wt[j * GN_LDW + i] = W[idx];
    }
    __syncthreads();

    int wave = tid >> 5;
    int lane = tid & 31;
    int m16  = lane & 15;        // row-within-strip (A) / col-within-tile (B, C/D)
    int half = lane >> 4;        // 0: K{0,1}, 1: K{2,3}
    int kb   = half * 2;

    int row0 = blockIdx.x * 128 + wave * 16;   // strip base (N is a multiple of 16 -> uniform)
    if (row0 >= n_nodes) return;               // wave-uniform: EXEC stays all-1s for WMMA

    int row = row0 + m16;
    float nd = norm_dst[row];
    const float* __restrict__ arow = agg + (size_t)row * GN_D;

    v8f acc0 = {}, acc1 = {}, acc2 = {}, acc3 = {};

    #pragma unroll
    for (int kk = 0; kk < 16; ++kk) {          // K = 64 in steps of 4
        int k = kk * 4 + kb;
        v2f a;
        a.x = arow[k]     * nd;                // fused norm_dst scaling
        a.y = arow[k + 1] * nd;

        v2f b0 = *(const v2f*)&Wt[( 0 + m16) * GN_LDW + k];
        v2f b1 = *(const v2f*)&Wt[(16 + m16) * GN_LDW + k];
        v2f b2 = *(const v2f*)&Wt[(32 + m16) * GN_LDW + k];
        v2f b3 = *(const v2f*)&Wt[(48 + m16) * GN_LDW + k];

        acc0 = __builtin_amdgcn_wmma_f32_16x16x4_f32(false, a, false, b0, (short)0, acc0, false, false);
        acc1 = __builtin_amdgcn_wmma_f32_16x16x4_f32(false, a, false, b1, (short)0, acc1, false, false);
        acc2 = __builtin_amdgcn_wmma_f32_16x16x4_f32(false, a, false, b2, (short)0, acc2, false, false);
        acc3 = __builtin_amdgcn_wmma_f32_16x16x4_f32(false, a, false, b3, (short)0, acc3, false, false);
    }

    float bv0 = bias[ 0 + m16];
    float bv1 = bias[16 + m16];
    float bv2 = bias[32 + m16];
    float bv3 = bias[48 + m16];

    // C/D layout: vgpr r holds row (r + 8*half), col = m16
    #pragma unroll
    for (int r = 0; r < 8; ++r) {
        int m = r + half * 8;
        float* o = out + (size_t)(row0 + m) * GN_D + m16;
        o[ 0] = acc0[r] + bv0;
        o[16] = acc1[r] + bv1;
        o[32] = acc2[r] + bv2;
        o[48] = acc3[r] + bv3;
    }
}

extern "C" void kernel_launch(void* const* d_in, const int* in_sizes, int n_in,
                              void* d_out, int out_size, void* d_ws, size_t ws_size,
                              hipStream_t stream) {
    const float* x   = (const float*)d_in[0];
    const int*   src = (const int*)  d_in[1];
    const int*   dst = (const int*)  d_in[2];
    const float* W   = (const float*)d_in[3];
    const float* b   = (const float*)d_in[4];
    float*       out = (float*)d_out;

    const int N = in_sizes[0] / GN_D;   // 50000
    const int E = in_sizes[1];          // 800000

    float* ws      = (float*)d_ws;
    float* deg_out = ws;                         // [N]  -> becomes norm_src
    float* deg_in  = ws + (size_t)N;             // [N]  -> becomes norm_dst
    float* agg     = ws + (size_t)2 * N;         // [N * 64]

    // 1) zero degrees + accumulator
    long long nzero = (long long)N * (GN_D + 2);
    int zb = (int)((nzero + 255) / 256); if (zb > 8192) zb = 8192;
    gn_zero<<<zb, 256, 0, stream>>>(ws, nzero);

    // 2) degree histograms
    gn_degree<<<(E + 255) / 256, 256, 0, stream>>>(src, dst, deg_out, deg_in, E);

    // 3) norms in place over both arrays (contiguous 2N floats)
    gn_norm<<<(2 * N + 255) / 256, 256, 0, stream>>>(ws, 2 * N);

    // 4) gather * norm_src, scatter-add into agg
    long long sthreads = (long long)E * 16;
    gn_scatter<<<(int)((sthreads + 255) / 256), 256, 0, stream>>>(x, src, dst, deg_out, agg, E);

    // 5) fused norm_dst * agg @ W + b via f32 WMMA
    gn_gemm_wmma<<<(N + 127) / 128, 256, 0, stream>>>(agg, deg_in, W, b, out, N);
}